// GeometricReconstructionLoss_77051713290714
// MI455X (gfx1250) — compile-verified
//
#include <hip/hip_runtime.h>
#include <hip/hip_bf16.h>
#include <math.h>

// ---------------------------------------------------------------------------
// GeometricReconstructionLoss for MI455X (gfx1250, wave32).
//
// Chamfer cross term is a K=3 GEMM -> pad K to 4 and use
// V_WMMA_F32_16X16X4_F32 with
//     A[m]   = (-2*q0, -2*q1, -2*q2, 1)          (query points)
//     B[:,n] = (x0, x1, x2, x0^2+x1^2+x2^2)      (database points)
// so D[m][n] = ||x_n||^2 - 2*q_m.x_n, whose argmin over n equals the true
// distance argmin (||q_m||^2 is constant along the reduced axis).
// One WMMA = a 16x16 tile of distance surrogates (1024 MACs).
//
// This revision makes the per-iteration A/B tile construction branchless
// (v_cndmask instead of EXEC-mask branches) so the hot loop is pure
// load -> select -> WMMA -> argmin-update.
// ---------------------------------------------------------------------------

typedef float v2f __attribute__((ext_vector_type(2)));
typedef float v8f __attribute__((ext_vector_type(8)));

#define NSETS 16   // B*S = 2*8
#define NPTS  2048
#define BDIM  2

__device__ __forceinline__ float smooth_l1(float x) {
    float d = fabsf(x);
    return (d < 1.0f) ? 0.5f * d * d : d - 0.5f;
}

// For each query point in Q[set], find argmin over DB[set] of
// ||db||^2 - 2*q.db  (== distance argmin). One wave handles a 16-row tile
// of queries and sweeps all 128 16-column database tiles with WMMA.
__global__ __launch_bounds__(256) void nn_argmin_wmma(
    const float* __restrict__ Q,    // [NSETS][NPTS][3]
    const float* __restrict__ DB,   // [NSETS][NPTS][3]
    int* __restrict__ out_idx)      // [NSETS][NPTS]
{
    const int lane  = threadIdx.x & 31;
    const int wave  = threadIdx.x >> 5;
    const int gwave = blockIdx.x * (blockDim.x >> 5) + wave;   // 0..2047
    const int ntiles = NPTS >> 4;                               // 128
    const int set   = gwave / ntiles;                           // wave-uniform
    const int mtile = gwave % ntiles;

    const int  h  = lane & 15;        // column/row selector within the tile
    const bool hi = lane >= 16;

    // --- A tile (16x4 f32, 2 VGPRs): lanes 0-15 carry K=0,1; 16-31 carry K=2,3.
    // Branchless: all lanes load their query point, then cndmask-select.
    const float* qp = Q + ((size_t)set * NPTS + (size_t)mtile * 16 + h) * 3;
    float q0 = qp[0], q1 = qp[1], q2 = qp[2];
    v2f A;
    A.x = hi ? (-2.0f * q2) : (-2.0f * q0);
    A.y = hi ? 1.0f         : (-2.0f * q1);

    // Running (value, index) argmin; lane holds 8 rows (M = r + 8*hi) at
    // column residue h; sweep covers n = nt*16 + h.
    float minv[8];
    int   mini[8];
#pragma unroll
    for (int r = 0; r < 8; ++r) { minv[r] = INFINITY; mini[r] = 0; }

    const float* dbase = DB + (size_t)set * NPTS * 3;
    for (int nt = 0; nt < ntiles; ++nt) {
        // --- B tile (4x16 f32, 2 VGPRs): VGPR0 = rows K=0 (lanes 0-15) and
        // K=2 (lanes 16-31); VGPR1 = rows K=1 and K=3 (||x||^2 pad row).
        // Branchless build: every lane loads all 3 coords + squared norm.
        const float* p = dbase + ((size_t)nt * 16 + h) * 3;
        float x0 = p[0], x1 = p[1], x2 = p[2];
        float sq = x0 * x0 + x1 * x1 + x2 * x2;
        v2f Bv;
        Bv.x = hi ? x2 : x0;
        Bv.y = hi ? sq : x1;

        v8f c = {0.f, 0.f, 0.f, 0.f, 0.f, 0.f, 0.f, 0.f};
        // D[m][n] = ||x_n||^2 - 2*q_m.x_n   (16x16 f32 in 8 VGPRs)
        v8f d = __builtin_amdgcn_wmma_f32_16x16x4_f32(
            /*neg_a=*/false, A, /*neg_b=*/false, Bv,
            /*c_mod=*/(short)0, c, /*reuse_a=*/false, /*reuse_b=*/false);

        const int nidx = nt * 16 + h;
#pragma unroll
        for (int r = 0; r < 8; ++r) {
            float v = d[r];
            if (v < minv[r]) { minv[r] = v; mini[r] = nidx; }  // strict <: first-min
        }
    }

    // Cross-lane reduce over the 16 column residues (xor masks 1..8 keep the
    // two 16-lane halves — which own disjoint row groups — separate).
#pragma unroll
    for (int off = 1; off < 16; off <<= 1) {
#pragma unroll
        for (int r = 0; r < 8; ++r) {
            float ov = __shfl_xor(minv[r], off, 32);
            int   oi = __shfl_xor(mini[r], off, 32);
            bool take = (ov < minv[r]) | ((ov == minv[r]) & (oi < mini[r]));
            minv[r] = take ? ov : minv[r];
            mini[r] = take ? oi : mini[r];                     // tie -> smaller n
        }
    }

    if (h == 0) {  // lane 0 writes rows 0..7 of the tile, lane 16 rows 8..15
        int* o = out_idx + set * NPTS + mtile * 16 + (hi ? 8 : 0);
#pragma unroll
        for (int r = 0; r < 8; ++r) o[r] = mini[r];
    }
}

__device__ float block_reduce_sum(float v, float* sm) {
    const int t = threadIdx.x;
    sm[t] = v;
    __syncthreads();
    for (int s = blockDim.x >> 1; s > 0; s >>= 1) {   // fixed order: deterministic
        if (t < s) sm[t] += sm[t + s];
        __syncthreads();
    }
    float r = sm[0];
    __syncthreads();
    return r;
}

// One block per set: smooth-L1 Chamfer terms + centroid term -> per-set partials.
__global__ __launch_bounds__(256) void chamfer_reduce(
    const float* __restrict__ X, const float* __restrict__ T,
    const float* __restrict__ w,
    const int* __restrict__ idxA,   // per pred: nearest target
    const int* __restrict__ idxB,   // per target: nearest pred
    float* __restrict__ partial)    // [NSETS][2]
{
    __shared__ float sm[256];
    const int set = blockIdx.x;
    const float* Xs = X + (size_t)set * NPTS * 3;
    const float* Ts = T + (size_t)set * NPTS * 3;
    const int* ia = idxA + set * NPTS;
    const int* ib = idxB + set * NPTS;

    float s1 = 0.f, s2 = 0.f;
    float cpx = 0.f, cpy = 0.f, cpz = 0.f;
    float ctx = 0.f, cty = 0.f, ctz = 0.f;

    for (int n = threadIdx.x; n < NPTS; n += blockDim.x) {
        float x0 = Xs[n * 3], x1 = Xs[n * 3 + 1], x2 = Xs[n * 3 + 2];
        float u0 = Ts[n * 3], u1 = Ts[n * 3 + 1], u2 = Ts[n * 3 + 2];
        int j = ia[n];
        float t0 = Ts[j * 3], t1 = Ts[j * 3 + 1], t2 = Ts[j * 3 + 2];
        s1 += smooth_l1(x0 - t0) + smooth_l1(x1 - t1) + smooth_l1(x2 - t2);
        int k = ib[n];
        float y0 = Xs[k * 3], y1 = Xs[k * 3 + 1], y2 = Xs[k * 3 + 2];
        s2 += smooth_l1(u0 - y0) + smooth_l1(u1 - y1) + smooth_l1(u2 - y2);
        cpx += x0; cpy += x1; cpz += x2;
        ctx += u0; cty += u1; ctz += u2;
    }

    s1  = block_reduce_sum(s1, sm);
    s2  = block_reduce_sum(s2, sm);
    cpx = block_reduce_sum(cpx, sm);
    cpy = block_reduce_sum(cpy, sm);
    cpz = block_reduce_sum(cpz, sm);
    ctx = block_reduce_sum(ctx, sm);
    cty = block_reduce_sum(cty, sm);
    ctz = block_reduce_sum(ctz, sm);

    if (threadIdx.x == 0) {
        const float invN3 = 1.0f / (float)(NPTS * 3);
        float per_set = (s1 + s2) * invN3;
        partial[set * 2 + 0] = per_set * w[set];
        const float invN = 1.0f / (float)NPTS;
        float dx = cpx * invN - ctx * invN;
        float dy = cpy * invN - cty * invN;
        float dz = cpz * invN - ctz * invN;
        partial[set * 2 + 1] = smooth_l1(dx) + smooth_l1(dy) + smooth_l1(dz);
    }
}

__global__ void finalize_kernel(const float* __restrict__ partial,
                                float* __restrict__ out)
{
    if (threadIdx.x == 0 && blockIdx.x == 0) {
        float loss = 0.f, lc = 0.f;
        for (int i = 0; i < NSETS; ++i) {       // fixed order: deterministic
            loss += partial[i * 2 + 0];
            lc   += partial[i * 2 + 1];
        }
        out[0] = loss / (float)BDIM;
        out[1] = lc / (float)(BDIM * 3);
    }
}

extern "C" void kernel_launch(void* const* d_in, const int* in_sizes, int n_in,
                              void* d_out, int out_size, void* d_ws, size_t ws_size,
                              hipStream_t stream) {
    const float* X = (const float*)d_in[0];        // [2,8,2048,3]
    const float* T = (const float*)d_in[1];        // [2,8,2048,3]
    const float* w = (const float*)d_in[2];        // [2,8] -> flat [16]
    float* out = (float*)d_out;                    // (loss, lossc)

    char* ws = (char*)d_ws;
    int*   idxA    = (int*)ws;                                   // 16*2048 ints
    int*   idxB    = (int*)(ws + (size_t)NSETS * NPTS * 4);      // 16*2048 ints
    float* partial = (float*)(ws + (size_t)2 * NSETS * NPTS * 4);// 32 floats

    // 16 sets * 128 m-tiles = 2048 waves; 8 waves/block -> 256 blocks (exact,
    // so every wave runs with full EXEC as WMMA requires).
    const int waves  = NSETS * (NPTS / 16);
    const int blocks = waves / 8;

    // Direction 1: for each pred point, nearest target.
    nn_argmin_wmma<<<blocks, 256, 0, stream>>>(X, T, idxA);
    // Direction 2: for each target point, nearest pred.
    nn_argmin_wmma<<<blocks, 256, 0, stream>>>(T, X, idxB);

    chamfer_reduce<<<NSETS, 256, 0, stream>>>(X, T, w, idxA, idxB, partial);
    finalize_kernel<<<1, 32, 0, stream>>>(partial, out);
}